// KalmanFilter_56616258896445
// MI455X (gfx1250) — compile-verified
//
#include <hip/hip_runtime.h>
#include <hip/hip_bf16.h>
#include <math.h>

// ---------------------------------------------------------------------------
// Kalman filter, DX=DY=64, B=64, T=512.  Serial scan in ONE workgroup
// (16 waves = 512 threads, wave32), all matrices resident in LDS (CDNA5 WGP
// has 320KB).  All 64x64x64 f32 matmuls use V_WMMA_F32_16X16X4_F32.
// The 538MB broadcast covariance output is written by a separate
// bandwidth-bound kernel.
// ---------------------------------------------------------------------------

typedef float v2f __attribute__((ext_vector_type(2)));
typedef float v8f __attribute__((ext_vector_type(8)));

#define NT   512          // threads per block (16 waves)
#define LD   68           // padded LDS row stride (floats), 16B-aligned rows
#define MSZ  (64 * LD)    // floats per 64x64 LDS matrix buffer (4352)
#define NBUF 14
#define TAIL (NBUF * MSZ) // small vectors/scalars region

// -32 * ln(2*pi)  ==  -DY * 0.5 * log(2*pi)
#define LLC  (-58.8120661251f)

// ---------------------------------------------------------------------------
// D = op(A) * op(B) [+ C], all 64x64 f32 in LDS with stride LD.
// 16 waves: wave w owns output tile (w>>2, w&3).  Per K-block of 4 we issue
// one v_wmma_f32_16x16x4_f32.  Layouts per CDNA5 ISA 7.12.2:
//   A 16x4 : lane L holds A[L%16, k0+v]   with k0 = 4*kk + 2*(L>>4), v=0,1
//   B 4x16 : lane L holds B[k0+v, L%16]
//   C/D    : vgpr v: lanes0-15 -> M=v, lanes16-31 -> M=v+8 ; N = L%16
// Forceinline so InferAddressSpaces promotes all pointers to LDS (no generic
// null-checks / 64-bit address math).  Leading __syncthreads() fences
// producers; dst must not alias A or B (dst==C is allowed: tile ownership is
// exclusive).
// ---------------------------------------------------------------------------
template <bool TA, bool TB, bool HASC>
__device__ __forceinline__ void mm64(float* __restrict__ D,
                                     const float* __restrict__ A,
                                     const float* __restrict__ B,
                                     const float* __restrict__ C) {
  __syncthreads();
  const int lane = threadIdx.x & 31;
  const int wave = threadIdx.x >> 5;          // 0..15
  const int ti = (wave >> 2) << 4;            // tile row origin
  const int tj = (wave & 3) << 4;             // tile col origin
  const int mr = lane & 15;
  const int hi = lane >> 4;                   // 0 or 1

  v8f acc;
  if (HASC) {
#pragma unroll
    for (int v = 0; v < 8; ++v) acc[v] = C[(ti + (hi << 3) + v) * LD + tj + mr];
  } else {
#pragma unroll
    for (int v = 0; v < 8; ++v) acc[v] = 0.0f;
  }

#pragma unroll
  for (int kk = 0; kk < 16; ++kk) {
    const int k0 = (kk << 2) + (hi << 1);
    v2f av, bv;
    if (!TA) {
      av[0] = A[(ti + mr) * LD + k0];
      av[1] = A[(ti + mr) * LD + k0 + 1];
    } else {
      av[0] = A[k0 * LD + ti + mr];
      av[1] = A[(k0 + 1) * LD + ti + mr];
    }
    if (!TB) {
      bv[0] = B[k0 * LD + tj + mr];
      bv[1] = B[(k0 + 1) * LD + tj + mr];
    } else {
      bv[0] = B[(tj + mr) * LD + k0];
      bv[1] = B[(tj + mr) * LD + k0 + 1];
    }
    acc = __builtin_amdgcn_wmma_f32_16x16x4_f32(false, av, false, bv,
                                                (short)0, acc, false, false);
  }

#pragma unroll
  for (int v = 0; v < 8; ++v) D[(ti + (hi << 3) + v) * LD + tj + mr] = acc[v];
}

// ---------------------------------------------------------------------------
// Main filter kernel: one block, persistent over the whole scan.
// ---------------------------------------------------------------------------
__global__ __launch_bounds__(NT) void kf_main(
    const float* __restrict__ obs,        // (T+1, 64, 64)
    const float* __restrict__ prior_mean, // (64)
    const float* __restrict__ prior_chol, // (64,64)
    const float* __restrict__ dyn_w,      // F (64,64)
    const float* __restrict__ dyn_b,      // (64)
    const float* __restrict__ dyn_chol,   // (64,64)
    const float* __restrict__ obs_w,      // H (64,64)
    const float* __restrict__ obs_b,      // (64)
    const float* __restrict__ obs_chol,   // (64,64)
    float* __restrict__ out_means,        // (T+1, 64, 64)
    float* __restrict__ out_covs,         // (T+1, 64, 64, 64) ; we fill b=0
    float* __restrict__ out_lls) {        // (T+1, 64)
  extern __shared__ float smem[];
  float* F   = smem + 0 * MSZ;   // dynamic_weight
  float* Hm  = smem + 1 * MSZ;   // observation_weight
  float* Qm  = smem + 2 * MSZ;   // process noise
  float* Rm  = smem + 3 * MSZ;   // observation noise
  float* Pst = smem + 4 * MSZ;   // state covariance
  float* Mst = smem + 5 * MSZ;   // state means (b, dx)
  float* S0  = smem + 6 * MSZ;   // pm
  float* S1  = smem + 7 * MSZ;   // T / HP / PmHt / Linv / Kt
  float* S2  = smem + 8 * MSZ;   // pP
  float* S3  = smem + 9 * MSZ;   // oC -> L -> Sinv -> t4n
  float* S4  = smem + 10 * MSZ;  // resT -> WIWn -> nm
  float* S5  = smem + 11 * MSZ;  // resb
  float* S6  = smem + 12 * MSZ;  // Z -> t3 -> newP
  float* Hn  = smem + 13 * MSZ;  // -H (precomputed, kills per-step negate)
  float* tail = smem + TAIL;     // [0:64) dyn_b, [64:128) obs_b,
                                 // [128:192) prior_mean, [192] logdet
  const int tid = threadIdx.x;

  // ---- preamble: load constants -------------------------------------------
  for (int idx = tid; idx < 4096; idx += NT) {
    const int r = idx >> 6, c = idx & 63;
    const float h = obs_w[idx];
    F[r * LD + c]  = dyn_w[idx];
    Hm[r * LD + c] = h;
    Hn[r * LD + c] = -h;
    S0[r * LD + c] = prior_chol[idx];
    S1[r * LD + c] = dyn_chol[idx];
    S3[r * LD + c] = obs_chol[idx];
  }
  if (tid < 64) {
    tail[tid]       = dyn_b[tid];
    tail[64 + tid]  = obs_b[tid];
    tail[128 + tid] = prior_mean[tid];
  }
  mm64<false, true, false>(Qm, S1, S1, nullptr);  // Q  = Ld Ld^T
  mm64<false, true, false>(Rm, S3, S3, nullptr);  // R  = Lo Lo^T
  mm64<false, true, false>(S2, S0, S0, nullptr);  // P0 = Lp Lp^T  (pP for t=0)

  // ---- serial scan --------------------------------------------------------
  for (int t = 0; t <= 512; ++t) {
    const float* y = obs + (size_t)t * 4096;

    if (t == 0) {
      __syncthreads();  // prior_chol reads done; safe to overwrite S0
      for (int idx = tid; idx < 4096; idx += NT)
        S0[(idx >> 6) * LD + (idx & 63)] = tail[128 + (idx & 63)];  // pm = prior_mean
    } else {
      mm64<false, true, false>(S0, Mst, F, nullptr);  // pm = m F^T
      __syncthreads();
      for (int idx = tid; idx < 4096; idx += NT)
        S0[(idx >> 6) * LD + (idx & 63)] += tail[idx & 63];         // + dyn_bias
      mm64<false, false, false>(S1, F, Pst, nullptr); // T  = F P
      mm64<false, true, true>(S2, S1, F, Qm);         // pP = T F^T + Q
    }

    mm64<false, false, false>(S1, Hm, S2, nullptr);   // HP   = H pP
    mm64<false, true, true>(S3, S1, Hm, Rm);          // oC   = HP H^T + R
    mm64<false, true, false>(S1, S0, Hm, nullptr);    // PmHt = pm H^T

    // residuals: resb (b,dy) and resT (dy,b)
    __syncthreads();
    for (int idx = tid; idx < 4096; idx += NT) {
      const int b = idx >> 6, i = idx & 63;
      const float rb = y[idx] - tail[64 + i];
      S5[b * LD + i] = rb;                       // y - obs_bias
      S4[i * LD + b] = rb - S1[b * LD + i];      // (y - om)^T
    }
    __syncthreads();

    // ---- Cholesky of oC (in place, lower triangle -> L) ----
    for (int j = 0; j < 64; ++j) {
      if (tid == 0) S3[j * LD + j] = sqrtf(S3[j * LD + j]);
      __syncthreads();
      const float dinv = 1.0f / S3[j * LD + j];
      if (tid > j && tid < 64) S3[tid * LD + j] *= dinv;
      __syncthreads();
      const int n = 63 - j;
      for (int idx = tid; idx < n * n; idx += NT) {
        const int i = j + 1 + idx / n;
        const int k = j + 1 + idx % n;
        if (k <= i) S3[i * LD + k] -= S3[i * LD + j] * S3[k * LD + j];
      }
      __syncthreads();
    }
    if (tid == 0) {
      float s = 0.0f;
      for (int j = 0; j < 64; ++j) s += logf(S3[j * LD + j]);
      tail[192] = 2.0f * s;
    }

    // ---- Linv = L^-1 (lower triangular) into S1 ----
    __syncthreads();
    for (int idx = tid; idx < 4096; idx += NT)
      S1[(idx >> 6) * LD + (idx & 63)] = 0.0f;
    __syncthreads();
    if (tid < 64) {
      const int c = tid;
      S1[c * LD + c] = 1.0f / S3[c * LD + c];
      for (int i = c + 1; i < 64; ++i) {
        float s = 0.0f;
        for (int k = c; k < i; ++k) s += S3[i * LD + k] * S1[k * LD + c];
        S1[i * LD + c] = -s / S3[i * LD + i];
      }
    }

    // ---- log-likelihood: Z = Linv res^T, q_b = ||Z[:,b]||^2 ----
    mm64<false, false, false>(S6, S1, S4, nullptr);   // Z
    __syncthreads();
    if (tid < 64) {
      float q = 0.0f;
      for (int i = 0; i < 64; ++i) {
        const float z = S6[i * LD + tid];
        q += z * z;
      }
      out_lls[t * 64 + tid] = LLC - 0.5f * tail[192] - 0.5f * q;
    }

    // ---- Bayes update ----
    mm64<true, false, false>(S3, S1, S1, nullptr);    // Sinv = Linv^T Linv
    mm64<false, false, false>(S6, S3, Hm, nullptr);   // t3   = Sinv H
    mm64<false, false, false>(S1, S6, S2, nullptr);   // Kt   = t3 pP   (= (CW Sinv)^T)
    mm64<true, false, false>(S4, Hn, S6, nullptr);    // WIWn = -(H^T t3)
    mm64<false, false, false>(S3, S4, S2, nullptr);   // t4n  = WIWn pP
    mm64<false, false, true>(S6, S2, S3, S2);         // newP = pP t4n + pP
    mm64<false, false, true>(S4, S0, S3, S0);         // nm   = pm t4n + pm
    mm64<false, false, true>(S4, S5, S1, S4);         // nm  += resb Kt

    // ---- commit state + outputs (cov only to batch slot 0) ----
    __syncthreads();
    for (int idx = tid; idx < 4096; idx += NT) {
      const int o = (idx >> 6) * LD + (idx & 63);
      const float nm = S4[o], np = S6[o];
      Mst[o] = nm;
      Pst[o] = np;
      out_means[(size_t)t * 4096 + idx] = nm;
      out_covs[(size_t)t * 64 * 4096 + idx] = np;
    }
    // next iteration's first mm64 carries the fencing barrier
  }
}

// ---------------------------------------------------------------------------
// Broadcast covs[t, 0, :, :] -> covs[t, b, :, :] for b = 1..63 (stream ~538MB)
// ---------------------------------------------------------------------------
__global__ void kf_bcast(float* __restrict__ covs) {
  const int bx = blockIdx.x;        // 0 .. 513*63-1
  const int t = bx / 63;
  const int b = 1 + bx % 63;
  const float* __restrict__ src = covs + (size_t)t * 64 * 4096;
  float* __restrict__ dst = covs + (size_t)t * 64 * 4096 + (size_t)b * 4096;
  for (int i = threadIdx.x; i < 4096; i += blockDim.x) dst[i] = src[i];
}

// ---------------------------------------------------------------------------
extern "C" void kernel_launch(void* const* d_in, const int* in_sizes, int n_in,
                              void* d_out, int out_size, void* d_ws,
                              size_t ws_size, hipStream_t stream) {
  (void)in_sizes; (void)n_in; (void)out_size; (void)d_ws; (void)ws_size;
  // setup_inputs order:
  // 0 time_extent, 1 observation, 2 prior_mean, 3 prior_chol, 4 dynamic_weight,
  // 5 dynamic_bias, 6 dynamic_chol, 7 observation_weight, 8 observation_bias,
  // 9 observation_chol
  const float* obs        = (const float*)d_in[1];
  const float* prior_mean = (const float*)d_in[2];
  const float* prior_chol = (const float*)d_in[3];
  const float* dyn_w      = (const float*)d_in[4];
  const float* dyn_b      = (const float*)d_in[5];
  const float* dyn_chol   = (const float*)d_in[6];
  const float* obs_w      = (const float*)d_in[7];
  const float* obs_b      = (const float*)d_in[8];
  const float* obs_chol   = (const float*)d_in[9];

  float* out       = (float*)d_out;
  float* out_means = out;                                  // 513*64*64
  float* out_covs  = out_means + (size_t)513 * 4096;       // 513*64*64*64
  float* out_lls   = out_covs + (size_t)513 * 64 * 4096;   // 513*64

  const size_t smem_bytes = (size_t)(NBUF * MSZ + 256) * sizeof(float);

  kf_main<<<1, NT, smem_bytes, stream>>>(obs, prior_mean, prior_chol, dyn_w,
                                         dyn_b, dyn_chol, obs_w, obs_b,
                                         obs_chol, out_means, out_covs,
                                         out_lls);
  kf_bcast<<<513 * 63, 256, 0, stream>>>(out_covs);
}